// SwiftTPConvolution_55018531062295
// MI455X (gfx1250) — compile-verified
//
#include <hip/hip_runtime.h>
#include <hip/hip_bf16.h>
#include <cstdint>

#define TPB   256
#define EPB   8            // edges per block
#define ROWX  256          // floats per x row   (4*MUL)
#define ROWW  320          // floats per weight row (5*MUL)
#define STAGE_F (ROWX + ROWW)          // 576 floats per staged edge
#define CHUNKS  (STAGE_F / 4)          // 144 b128 chunks

static_assert(STAGE_F % 4 == 0, "stage must be float4 multiple");
static_assert(CHUNKS <= TPB, "one chunk per thread");

// ---- CDNA5 async global->LDS path (guarded; falls back to sync copy) ----
#if defined(__AMDGCN__) && __has_builtin(__builtin_amdgcn_global_load_async_to_lds_b128) && __has_builtin(__builtin_amdgcn_s_wait_asynccnt)
#define USE_ASYNC 1
#else
#define USE_ASYNC 0
#endif

typedef int v4i __attribute__((ext_vector_type(4)));

__device__ __forceinline__ void stage_b128(const float* g, float* l) {
#if USE_ASYNC
  __builtin_amdgcn_global_load_async_to_lds_b128(
      (__attribute__((address_space(1))) v4i*)(g),
      (__attribute__((address_space(3))) v4i*)(l), 0, 0);
#else
  *reinterpret_cast<float4*>(l) = *reinterpret_cast<const float4*>(g);
#endif
}

#if USE_ASYNC
#define WAIT_ASYNC(n) __builtin_amdgcn_s_wait_asynccnt(n)
#else
#define WAIT_ASYNC(n)
#endif

// ---- native f32 atomic add (avoid CAS-loop lowering of atomicAdd) ----
__device__ __forceinline__ void atomic_add_f32(float* p, float v) {
#if defined(__AMDGCN__) && __has_builtin(__builtin_amdgcn_global_atomic_fadd_f32)
  __builtin_amdgcn_global_atomic_fadd_f32(
      (__attribute__((address_space(1))) float*)p, v);
#else
  atomicAdd(p, v);
#endif
}

// NOTE: main kernel first so the disasm snippet shows it.
__global__ void __launch_bounds__(TPB)
swift_tp_conv_kernel(const float* __restrict__ x,
                     const float* __restrict__ edge_vec,
                     const float* __restrict__ weight,
                     const int*   __restrict__ edge_src,
                     const int*   __restrict__ edge_dst,
                     float*       __restrict__ out,
                     int E) {
  // double-buffered staging: [x row (256f) | weight row (320f)] per edge
  __shared__ float sbuf[2][STAGE_F];

  const int t  = threadIdx.x;
  const int e0 = blockIdx.x * EPB;
  if (e0 >= E) return;
  const int eend = (e0 + EPB < E) ? (e0 + EPB) : E;

  // stage first edge
  {
    const int src = edge_src[e0];
    if (t < CHUNKS) {
      const float* g = (t < ROWX / 4)
                         ? (x + (size_t)src * ROWX + 4 * t)
                         : (weight + (size_t)e0 * ROWW + 4 * (t - ROWX / 4));
      stage_b128(g, &sbuf[0][4 * t]);
    }
  }

  const int u    = t & 63;   // channel
  const int comp = t >> 6;   // 0 = scalar msg, 1..3 = vector components (wave-uniform)

  const float SQ3     = 1.7320508075688772f;
  const float INV_SQ3 = 0.57735026918962576f;
  const float INV_SQ6 = 0.40824829046386302f;
  const float PW0c    = 0.70710678118654752f;

  for (int e = e0; e < eend; ++e) {
    const int cur = (e - e0) & 1;
    const int en  = e + 1;
    if (en < eend) {
      const int srcn = edge_src[en];
      if (t < CHUNKS) {
        const float* g = (t < ROWX / 4)
                           ? (x + (size_t)srcn * ROWX + 4 * t)
                           : (weight + (size_t)en * ROWW + 4 * (t - ROWX / 4));
        stage_b128(g, &sbuf[cur ^ 1][4 * t]);
      }
      WAIT_ASYNC(1);   // per-wave: exactly 1 async op per stage -> current stage landed
    } else {
      WAIT_ASYNC(0);   // drain on last iteration
    }
    __syncthreads();

    const float* S = sbuf[cur];

    // spherical harmonics l=1 (uniform per block; scalarizable loads)
    const float v0 = edge_vec[3 * (size_t)e + 0];
    const float v1 = edge_vec[3 * (size_t)e + 1];
    const float v2 = edge_vec[3 * (size_t)e + 2];
    const float nn  = sqrtf(v0 * v0 + v1 * v1 + v2 * v2);
    const float inv = 1.0f / fmaxf(nn, 1e-12f);
    const float s1x = SQ3 * v0 * inv;
    const float s1y = SQ3 * v1 * inv;
    const float s1z = SQ3 * v2 * inv;

    const float xs0 = S[u];
    const float a0  = S[64 + 3 * u + 0];
    const float a1  = S[64 + 3 * u + 1];
    const float a2  = S[64 + 3 * u + 2];

    const int dst = edge_dst[e];
    float* orow = out + (size_t)dst * ROWX;

    float msg;
    int   idx;
    if (comp == 0) {
      const float w000 = S[ROWX + 0 * 64 + u];
      const float w110 = S[ROWX + 3 * 64 + u];
      const float dot  = a0 * s1x + a1 * s1y + a2 * s1z;
      msg = PW0c * (w000 * xs0 + w110 * dot * INV_SQ3);
      idx = u;
    } else {
      const int k = comp - 1;
      const float w011 = S[ROWX + 1 * 64 + u];
      const float w101 = S[ROWX + 2 * 64 + u];
      const float w111 = S[ROWX + 4 * 64 + u];
      const float shk  = (k == 0) ? s1x : (k == 1) ? s1y : s1z;          // sh1[k]
      const float shk1 = (k == 0) ? s1y : (k == 1) ? s1z : s1x;          // sh1[(k+1)%3]
      const float shk2 = (k == 0) ? s1z : (k == 1) ? s1x : s1y;          // sh1[(k+2)%3]
      const float ak   = (k == 0) ? a0  : (k == 1) ? a1  : a2;
      const float ak1  = (k == 0) ? a1  : (k == 1) ? a2  : a0;
      const float ak2  = (k == 0) ? a2  : (k == 1) ? a0  : a1;
      const float cr   = ak1 * shk2 - ak2 * shk1;                        // cross(xs1, sh1)[k]
      msg = w011 * xs0 * shk * INV_SQ3 + w101 * ak * INV_SQ3 + w111 * cr * INV_SQ6;
      idx = 64 + 3 * u + k;
    }
    atomic_add_f32(orow + idx, msg);
    __syncthreads();   // buffer may be overwritten by next iteration's staging
  }
}

// simple, bloat-free zeroing: float4 body + tiny tail on block 0
__global__ void __launch_bounds__(TPB)
zero_f32_kernel(float* __restrict__ p, int n4, int ntail) {
  const int i = blockIdx.x * TPB + threadIdx.x;
  if (i < n4) {
    float4 z; z.x = 0.f; z.y = 0.f; z.z = 0.f; z.w = 0.f;
    *reinterpret_cast<float4*>(p + 4 * (size_t)i) = z;
  }
  if (blockIdx.x == 0 && threadIdx.x < ntail) {
    p[4 * (size_t)n4 + threadIdx.x] = 0.f;
  }
}

extern "C" void kernel_launch(void* const* d_in, const int* in_sizes, int n_in,
                              void* d_out, int out_size, void* d_ws, size_t ws_size,
                              hipStream_t stream) {
  const float* x        = (const float*)d_in[0];
  const float* edge_vec = (const float*)d_in[1];
  const float* weight   = (const float*)d_in[2];
  const int*   edge_src = (const int*)d_in[3];
  const int*   edge_dst = (const int*)d_in[4];
  float* out = (float*)d_out;

  const int E = in_sizes[1] / 3;

  // zero the accumulator (d_out is poisoned before timing)
  const int n4    = out_size / 4;
  const int ntail = out_size - 4 * n4;
  const int zblocks = (n4 + TPB - 1) / TPB;
  zero_f32_kernel<<<(zblocks > 0 ? zblocks : 1), TPB, 0, stream>>>(out, n4, ntail);

  const int blocks = (E + EPB - 1) / EPB;
  swift_tp_conv_kernel<<<blocks, TPB, 0, stream>>>(x, edge_vec, weight,
                                                   edge_src, edge_dst, out, E);
}